// DTI_TS_59889023975768
// MI455X (gfx1250) — compile-verified
//
#include <hip/hip_runtime.h>

// Problem constants (match reference)
#define B_N 512
#define T_N 365
#define C_N 10
#define P_N 64
#define K1  (T_N * C_N)   // 3650 = cross-term K
#define QS  368           // padded K for the p2 phase (mult of 4)

typedef __attribute__((ext_vector_type(2))) float v2f;
typedef __attribute__((ext_vector_type(8))) float v8f;

// d_out layout (float elements), tuple flattened in return order
#define O_OUT   0
#define O_IN    (B_N * T_N * C_N)
#define O_DIST  (2 * B_N * T_N * C_N)
#define O_IDX   (O_DIST + B_N * P_N)
#define O_LAB   (O_IDX + B_N)
#define O_MASK  (O_LAB + B_N)

// ---------------------------------------------------------------------------
// Kernel 1: Q[p,t] = sum_c proto[p,t,c]^2, zero-padded to stride QS
// ---------------------------------------------------------------------------
__global__ __launch_bounds__(256) void dti_prep_q(const float* __restrict__ Pr,
                                                  float* __restrict__ Q) {
    int i = blockIdx.x * 256 + threadIdx.x;       // 0 .. P_N*QS-1 (grid sized exactly)
    int p = i / QS, t = i % QS;
    float q = 0.f;
    if (t < T_N) {
        const float* r = Pr + p * K1 + t * C_N;
#pragma unroll
        for (int c = 0; c < C_N; ++c) q += r[c] * r[c];
    }
    Q[i] = q;
}

// ---------------------------------------------------------------------------
// Kernel 2: distances via V_WMMA_F32_16X16X4_F32.
//   grid.x = B/16 = 32 blocks; block = 128 threads = 4 waves (wave32).
//   Wave w computes the 16x16 tile (b-rows of this block) x (protos w*16..w*16+15).
//   Phase 1 (K=3650): A = -2*mask*x fragment, B = proto fragment; also
//   accumulates 4*x2 per lane. Phase 2 (K=368): A = mask, B = Q (padded).
// ---------------------------------------------------------------------------
__global__ __launch_bounds__(128) void dti_dist_gemm(
    const float* __restrict__ X,    // [B, 3650]
    const float* __restrict__ Mk,   // [B, 365]
    const float* __restrict__ Pr,   // [P, 3650]
    const float* __restrict__ Q,    // [P, QS] (ws)
    float* __restrict__ Dist)       // [B, P]  (d_out + O_DIST)
{
    const int lane = threadIdx.x & 31;
    const int wave = threadIdx.x >> 5;
    const int half = lane >> 4;          // 0: K+0/K+1, 1: K+2/K+3 (ISA A/B layout)
    const int mrow = lane & 15;          // A row within M-tile == B column
    const int brow = blockIdx.x * 16 + mrow;   // sample row for A loads
    const int prow = wave * 16 + mrow;         // prototype row for B loads

    const float* Arow = X  + (size_t)brow * K1;
    const float* MRow = Mk + (size_t)brow * T_N;
    const float* Brow = Pr + (size_t)prow * K1;
    const float* QRow = Q  + (size_t)prow * QS;

    __builtin_prefetch(Arow, 0, 3);
    __builtin_prefetch(Brow, 0, 3);

    v8f acc = {};                        //  -2*xp + p2
    float sq = 0.f;                      //  4 * partial ||mask*x||^2

    // ---- Phase 1: cross term, K = 3650 = 912*4 + 2 ------------------------
    for (int k0 = 0; k0 < 3648; k0 += 4) {
        const int kb = k0 + 2 * half;                       // 8B-aligned offset
        v2f xv = *(const v2f*)(Arow + kb);                  // global_load_b64
        float m0 = MRow[kb / C_N];
        float m1 = MRow[(kb + 1) / C_N];
        v2f a; a.x = -2.f * xv.x * m0; a.y = -2.f * xv.y * m1;
        sq += a.x * a.x + a.y * a.y;                        // = 4*(m*x)^2
        v2f b = *(const v2f*)(Brow + kb);                   // global_load_b64
        acc = __builtin_amdgcn_wmma_f32_16x16x4_f32(
                  false, a, false, b, (short)0, acc, false, false);
    }
    { // tail: k = 3648..3651; lanes with half==1 cover k>=3650 -> zero
        const int kb = 3648 + 2 * half;
        v2f a = {0.f, 0.f}, b = {0.f, 0.f};
        if (kb < K1) {                                      // kb==3648 (+1 = 3649 ok)
            v2f xv = *(const v2f*)(Arow + kb);
            a.x = -2.f * xv.x * MRow[kb / C_N];
            a.y = -2.f * xv.y * MRow[(kb + 1) / C_N];
            sq += a.x * a.x + a.y * a.y;
            b = *(const v2f*)(Brow + kb);
        }
        acc = __builtin_amdgcn_wmma_f32_16x16x4_f32(
                  false, a, false, b, (short)0, acc, false, false);
    }

    // ---- Phase 2: p2 term, K = 365 padded to 368 --------------------------
    for (int k0 = 0; k0 < 364; k0 += 4) {                   // k <= 363: no guards
        const int kb = k0 + 2 * half;
        v2f a; a.x = MRow[kb]; a.y = MRow[kb + 1];          // mask row (4B aligned)
        v2f b = *(const v2f*)(QRow + kb);                   // QS stride -> 8B aligned
        acc = __builtin_amdgcn_wmma_f32_16x16x4_f32(
                  false, a, false, b, (short)0, acc, false, false);
    }
    { // tail: k = 364..367; only k==364 is live (Q padding already zero)
        const int kb = 364 + 2 * half;
        v2f a = {0.f, 0.f}, b = {0.f, 0.f};
        if (kb < T_N) { a.x = MRow[kb]; b.x = QRow[kb]; }
        acc = __builtin_amdgcn_wmma_f32_16x16x4_f32(
                  false, a, false, b, (short)0, acc, false, false);
    }

    // ---- Epilogue: add x2[row], write distances ---------------------------
    __shared__ float sqb[4][32];
    sqb[wave][lane] = sq;
    __syncthreads();

    const int col = lane & 15;                              // prototype within tile
#pragma unroll
    for (int i = 0; i < 8; ++i) {
        const int row = i + 8 * half;                       // ISA C/D layout
        const float x2 = 0.25f * (sqb[wave][row] + sqb[wave][row + 16]);
        const int b = blockIdx.x * 16 + row;
        Dist[(size_t)b * P_N + wave * 16 + col] = acc[i] + x2;
    }
}

// ---------------------------------------------------------------------------
// Kernel 3: per-sample argmin over 64 distances (wave32 butterfly, first-index
// tie break like jnp.argmin) + emit indices/labels as float.
// ---------------------------------------------------------------------------
__global__ __launch_bounds__(256) void dti_argmin(
    const float* __restrict__ Dist, const int* __restrict__ label,
    float* __restrict__ out_idx, float* __restrict__ out_lab,
    int* __restrict__ ws_idx)
{
    const int lane = threadIdx.x & 31;
    const int wave = threadIdx.x >> 5;
    const int b = blockIdx.x * 8 + wave;

    float v0 = Dist[(size_t)b * P_N + lane];
    float v1 = Dist[(size_t)b * P_N + 32 + lane];
    float val; int idx;
    if (v1 < v0) { val = v1; idx = lane + 32; } else { val = v0; idx = lane; }

#pragma unroll
    for (int off = 16; off >= 1; off >>= 1) {
        float ov = __shfl_xor(val, off, 32);
        int   oi = __shfl_xor(idx, off, 32);
        if (ov < val || (ov == val && oi < idx)) { val = ov; idx = oi; }
    }
    if (lane == 0) {
        ws_idx[b]  = idx;
        out_idx[b] = (float)idx;
        out_lab[b] = (float)label[b];
    }
}

// ---------------------------------------------------------------------------
// Kernel 4: output_seq[b,:] = prototypes[idx[b], :]
// ---------------------------------------------------------------------------
__global__ __launch_bounds__(256) void dti_gather(
    const float* __restrict__ Pr, const int* __restrict__ ws_idx,
    float* __restrict__ outseq)
{
    const int r = blockIdx.x * 256 + threadIdx.x;   // 0..K1-1 (+ tail guard)
    const int b = blockIdx.y;
    if (r < K1)
        outseq[(size_t)b * K1 + r] = Pr[(size_t)ws_idx[b] * K1 + r];
}

// ---------------------------------------------------------------------------
extern "C" void kernel_launch(void* const* d_in, const int* in_sizes, int n_in,
                              void* d_out, int out_size, void* d_ws, size_t ws_size,
                              hipStream_t stream) {
    const float* x     = (const float*)d_in[0];   // [B,T,C]
    const float* mask  = (const float*)d_in[1];   // [B,T]
    const int*   label = (const int*)  d_in[2];   // [B]
    const float* proto = (const float*)d_in[3];   // [P,T,C]
    float* out = (float*)d_out;

    float* Q      = (float*)d_ws;                                   // [P, QS]
    int*   ws_idx = (int*)((char*)d_ws + P_N * QS * sizeof(float)); // [B]

    // Pass-through outputs: input_seq and mask copies (D2D, capture-safe)
    hipMemcpyAsync(out + O_IN,  x,    (size_t)B_N * K1  * sizeof(float),
                   hipMemcpyDeviceToDevice, stream);
    hipMemcpyAsync(out + O_MASK, mask, (size_t)B_N * T_N * sizeof(float),
                   hipMemcpyDeviceToDevice, stream);

    dti_prep_q<<<(P_N * QS) / 256, 256, 0, stream>>>(proto, Q);
    dti_dist_gemm<<<B_N / 16, 128, 0, stream>>>(x, mask, proto, Q, out + O_DIST);
    dti_argmin<<<B_N / 8, 256, 0, stream>>>(out + O_DIST, label,
                                            out + O_IDX, out + O_LAB, ws_idx);
    dti_gather<<<dim3((K1 + 255) / 256, B_N), 256, 0, stream>>>(proto, ws_idx,
                                                                out + O_OUT);
}